// AngularBasis_91027536872035
// MI455X (gfx1250) — compile-verified
//
#include <hip/hip_runtime.h>
#include <math.h>

// ---------------------------------------------------------------------------
// Real spherical harmonics Y_{l,m}, l<=8, output [N, 81] f32.
// Strategy: memory-bound (324B out / 8B in per point, ~28.5us at 23.3TB/s).
// Compute in registers with fully-unrolled Legendre + Chebyshev recurrences,
// stage each wave's 32x81 tile in LDS (stride-81 = odd => bank-conflict-free),
// then stream the contiguous 10368B tile to HBM with coalesced CDNA5
// global_store_async_from_lds_b128 (ASYNCcnt path), non-temporal hint.
// ---------------------------------------------------------------------------

#define LMAX  8
#define NCOEF 81                 // (LMAX+1)^2
#define WAVE  32
#define WAVES_PER_BLOCK 4
#define BLOCK (WAVE * WAVES_PER_BLOCK)
#define TILE_DW (WAVE * NCOEF)   // 2592 dwords (10368 B) per wave tile

// ---- compile-time normalization table K(l,m) (with sqrt(2) folded for m>0) --
constexpr double csqrt_(double x) {
    double g = x > 1.0 ? x : 1.0;
    for (int i = 0; i < 64; ++i) g = 0.5 * (g + x / g);
    return g;
}
constexpr double dfact_(int n) {
    double r = 1.0;
    for (int i = 2; i <= n; ++i) r *= (double)i;
    return r;
}
struct KT { float k[LMAX + 1][LMAX + 1]; };
constexpr KT makeK() {
    KT t{};
    for (int l = 0; l <= LMAX; ++l)
        for (int m = 0; m <= l; ++m) {
            double v = csqrt_((2.0 * l + 1.0) / (4.0 * 3.14159265358979323846)
                              * dfact_(l - m) / dfact_(l + m));
            if (m > 0) v *= csqrt_(2.0);
            t.k[l][m] = (float)v;
        }
    return t;
}
constexpr KT KTab = makeK();

__global__ __launch_bounds__(BLOCK)
void AngularBasis_sh_kernel(const float* __restrict__ ct_in,
                            const float* __restrict__ phi_in,
                            float* __restrict__ out, int n)
{
    __shared__ float tile[WAVES_PER_BLOCK * TILE_DW];   // 41472 B

    const int lane = threadIdx.x & (WAVE - 1);
    const int w    = threadIdx.x >> 5;
    const int p    = blockIdx.x * BLOCK + threadIdx.x;
    const int waveStart = p - lane;                     // multiple of 32
    float* wtile = &tile[w * TILE_DW];

    if (p < n) {
        const float ct = ct_in[p];
        const float ph = phi_in[p];
        const float st = sqrtf(fmaxf(1.0f - ct * ct, 0.0f));

        // cos(m*phi), sin(m*phi) via Chebyshev recurrence
        float cm[LMAX + 1], sm[LMAX + 1];
        cm[0] = 1.0f; sm[0] = 0.0f;
        __sincosf(ph, &sm[1], &cm[1]);
        const float twoc = 2.0f * cm[1];
#pragma unroll
        for (int m = 2; m <= LMAX; ++m) {
            cm[m] = twoc * cm[m - 1] - cm[m - 2];
            sm[m] = twoc * sm[m - 1] - sm[m - 2];
        }

        float* yrow = wtile + lane * NCOEF;   // odd stride 81 -> no bank conflicts
        float pmm = 1.0f;                     // P_m^m running value
#pragma unroll
        for (int m = 0; m <= LMAX; ++m) {
            if (m > 0) pmm *= st * (float)(2 * m - 1);
            {   // (l = m, m)
                const float kp = KTab.k[m][m] * pmm;
                if (m == 0) yrow[0] = kp;
                else { yrow[m * m + 2 * m] = kp * cm[m]; yrow[m * m] = kp * sm[m]; }
            }
            if (m < LMAX) {
                float pl2 = pmm;
                float pl  = ct * (float)(2 * m + 1) * pmm;   // P_{m+1}^m
                {
                    const int l = m + 1;
                    const float kp = KTab.k[l][m] * pl;
                    if (m == 0) yrow[l * l + l] = kp;
                    else { yrow[l * l + l + m] = kp * cm[m]; yrow[l * l + l - m] = kp * sm[m]; }
                }
#pragma unroll
                for (int l = m + 2; l <= LMAX; ++l) {
                    const float a = (float)(2 * l - 1) / (float)(l - m);
                    const float b = (float)(l + m - 1) / (float)(l - m);
                    const float pn = a * ct * pl - b * pl2;
                    pl2 = pl; pl = pn;
                    const float kp = KTab.k[l][m] * pn;
                    if (m == 0) yrow[l * l + l] = kp;
                    else { yrow[l * l + l + m] = kp * cm[m]; yrow[l * l + l - m] = kp * sm[m]; }
                }
            }
        }
    }

    // All of this wave's LDS writes must land before the async engine reads LDS.
    asm volatile("s_wait_dscnt 0x0" ::: "memory");

    const int nvalid = n - waveStart < WAVE ? (n - waveStart) : WAVE;
    const unsigned long long gbase =
        (unsigned long long)(out + (size_t)waveStart * NCOEF);     // 16B aligned
    const unsigned lbase = (unsigned)(unsigned long long)(void*)wtile;

    if (nvalid == WAVE) {
        // 648 float4 per wave tile: fully coalesced async LDS->global stores
#pragma unroll 1
        for (int f = lane; f < TILE_DW / 4; f += WAVE) {
            unsigned long long ga = gbase + (unsigned long long)f * 16u;
            unsigned           la = lbase + (unsigned)f * 16u;
            asm volatile("global_store_async_from_lds_b128 %0, %1, off th:TH_STORE_NT"
                         :: "v"(ga), "v"(la) : "memory");
        }
    } else if (nvalid > 0) {
        // tail wave: dword-granular async stores over the valid region only
        const int ndw = nvalid * NCOEF;
#pragma unroll 1
        for (int f = lane; f < ndw; f += WAVE) {
            unsigned long long ga = gbase + (unsigned long long)f * 4u;
            unsigned           la = lbase + (unsigned)f * 4u;
            asm volatile("global_store_async_from_lds_b32 %0, %1, off th:TH_STORE_NT"
                         :: "v"(ga), "v"(la) : "memory");
        }
    }
    // S_ENDPGM performs an implicit wait-idle, draining ASYNCcnt before retire.
}

extern "C" void kernel_launch(void* const* d_in, const int* in_sizes, int n_in,
                              void* d_out, int out_size, void* d_ws, size_t ws_size,
                              hipStream_t stream) {
    // setup_inputs order: l_max (int scalar, ==8 compiled in), cos_theta, phi
    const float* ct  = (const float*)d_in[1];
    const float* ph  = (const float*)d_in[2];
    float*       out = (float*)d_out;
    const int n = in_sizes[1];
    const int blocks = (n + BLOCK - 1) / BLOCK;
    AngularBasis_sh_kernel<<<blocks, BLOCK, 0, stream>>>(ct, ph, out, n);
}